// SoftEmbeddedDecisionRules_78108275245686
// MI455X (gfx1250) — compile-verified
//
#include <hip/hip_runtime.h>
#include <stdint.h>

#define LOG2E_F 1.44269504088896340736f

__device__ __forceinline__ float fast_rcp(float x)  { return __builtin_amdgcn_rcpf(x); }
__device__ __forceinline__ float fast_exp2(float x) { return __builtin_amdgcn_exp2f(x); }

// xor-lane shuffle within wave32 via ds_swizzle (offset[15]=0 group-of-32 mode:
// and_mask=0x1F, or_mask=0, xor_mask=XORM)
template <int XORM>
__device__ __forceinline__ float swz_xor(float x) {
  return __int_as_float(
      __builtin_amdgcn_ds_swizzle(__float_as_int(x), (XORM << 10) | 0x1F));
}

// softmax prob of own child: 1 / (1 + exp((sib - self)/blk)); s = log2(e)/blk
__device__ __forceinline__ float own_prob(float self, float sib, float s) {
  return fast_rcp(1.0f + fast_exp2((sib - self) * s));
}

// both children probs of one node with child sums A (left), B (right); s = log2(e)/blk
__device__ __forceinline__ void pair_probs(float A, float B, float s,
                                           float& pL, float& pR) {
  float e = fast_exp2((B - A) * s);
  float r = fast_rcp(1.0f + e);
  pL = r;
  pR = e * r;
}

__global__ __launch_bounds__(256) void nbdt_rules_kernel(
    const float* __restrict__ in, float* __restrict__ out, int nrows) {
  __shared__ __align__(16) float sh[8 * 1024];  // 8 rows * 4KB = 32KB
  const int lane = threadIdx.x & 31;
  const int wave = threadIdx.x >> 5;
  const int row  = blockIdx.x * 8 + wave;
  if (row >= nrows) return;

  float* wsh = &sh[wave * 1024];
  const uint64_t gbase = (uint64_t)(uintptr_t)in + (uint64_t)row * 4096ull;

  // ---- async global -> LDS staging (CDNA5 async-tensor path) ----
  // Writer: instruction j moves chunks c = j*32+lane (16B each): global side is
  // one contiguous 512B burst per instruction. LDS side is XOR-swizzled so that
  // the reader phase (lane reads chunks c = lane*8+i) is bank-spread.
  #pragma unroll
  for (int j = 0; j < 8; ++j) {
    uint32_t c  = (uint32_t)(j * 32 + lane);
    uint32_t p  = c ^ ((c >> 4) & 0xFu);              // bijective bank swizzle
    uint64_t ga = gbase + (uint64_t)c * 16ull;
    uint32_t la = (uint32_t)(uintptr_t)(&wsh[p * 4]); // LDS byte address
    asm volatile("global_load_async_to_lds_b128 %0, %1, off"
                 :: "v"(la), "v"(ga) : "memory");
  }
  asm volatile("s_wait_asynccnt 0" ::: "memory");

  // ---- lane reads its 32 contiguous classes [32*lane, 32*lane+32) ----
  float v[32];
  #pragma unroll
  for (int i = 0; i < 8; ++i) {
    uint32_t c = (uint32_t)(lane * 8 + i);
    uint32_t p = c ^ ((c >> 4) & 0xFu);
    const float4 q = *reinterpret_cast<const float4*>(&wsh[p * 4]);
    v[4 * i + 0] = q.x; v[4 * i + 1] = q.y;
    v[4 * i + 2] = q.z; v[4 * i + 3] = q.w;
  }

  // ---- in-register segment-tree sums (block sizes 2..32) ----
  float s2[16];
  #pragma unroll
  for (int k = 0; k < 16; ++k) s2[k] = v[2 * k] + v[2 * k + 1];
  float s4[8];
  #pragma unroll
  for (int k = 0; k < 8; ++k) s4[k] = s2[2 * k] + s2[2 * k + 1];
  float s8[4];
  #pragma unroll
  for (int k = 0; k < 4; ++k) s8[k] = s4[2 * k] + s4[2 * k + 1];
  float s16[2];
  s16[0] = s8[0] + s8[1];
  s16[1] = s8[2] + s8[3];
  const float s32v = s16[0] + s16[1];

  // ---- cross-lane sums (block sizes 64..1024) via ds_swizzle xor tree ----
  const float t64   = s32v + swz_xor<1>(s32v);
  const float t128  = t64  + swz_xor<2>(t64);
  const float t256  = t128 + swz_xor<4>(t128);
  const float t512  = t256 + swz_xor<8>(t256);
  const float t1024 = t512 + swz_xor<16>(t512);

  // ---- tree levels with block >= 32: one shared factor per lane ----
  float outer;
  outer  = own_prob(t512, t1024 - t512, LOG2E_F / 512.0f);  // d=0, blk=512
  outer *= own_prob(t256, t512  - t256, LOG2E_F / 256.0f);  // d=1
  outer *= own_prob(t128, t256  - t128, LOG2E_F / 128.0f);  // d=2
  outer *= own_prob(t64,  t128  - t64,  LOG2E_F / 64.0f);   // d=3
  outer *= own_prob(s32v, t64   - s32v, LOG2E_F / 32.0f);   // d=4

  // ---- tree levels inside the lane: cumulative-product down the tree ----
  float c16[2];
  {
    float pL, pR;
    pair_probs(s16[0], s16[1], LOG2E_F / 16.0f, pL, pR);    // d=5, blk=16
    c16[0] = outer * pL; c16[1] = outer * pR;
  }
  float c8v[4];
  #pragma unroll
  for (int k = 0; k < 2; ++k) {                             // d=6, blk=8
    float pL, pR;
    pair_probs(s8[2 * k], s8[2 * k + 1], LOG2E_F / 8.0f, pL, pR);
    c8v[2 * k] = c16[k] * pL; c8v[2 * k + 1] = c16[k] * pR;
  }
  float c4v[8];
  #pragma unroll
  for (int k = 0; k < 4; ++k) {                             // d=7, blk=4
    float pL, pR;
    pair_probs(s4[2 * k], s4[2 * k + 1], LOG2E_F / 4.0f, pL, pR);
    c4v[2 * k] = c8v[k] * pL; c4v[2 * k + 1] = c8v[k] * pR;
  }
  float c2v[16];
  #pragma unroll
  for (int k = 0; k < 8; ++k) {                             // d=8, blk=2
    float pL, pR;
    pair_probs(s2[2 * k], s2[2 * k + 1], LOG2E_F / 2.0f, pL, pR);
    c2v[2 * k] = c4v[k] * pL; c2v[2 * k + 1] = c4v[k] * pR;
  }
  float o[32];
  #pragma unroll
  for (int k = 0; k < 16; ++k) {                            // d=9, blk=1
    float pL, pR;
    pair_probs(v[2 * k], v[2 * k + 1], LOG2E_F, pL, pR);
    o[2 * k] = c2v[k] * pL; o[2 * k + 1] = c2v[k] * pR;
  }

  // ---- store 32 contiguous outputs as 8x b128 ----
  float4* op = reinterpret_cast<float4*>(out + (size_t)row * 1024 + lane * 32);
  #pragma unroll
  for (int i = 0; i < 8; ++i)
    op[i] = make_float4(o[4 * i + 0], o[4 * i + 1], o[4 * i + 2], o[4 * i + 3]);
}

extern "C" void kernel_launch(void* const* d_in, const int* in_sizes, int n_in,
                              void* d_out, int out_size, void* d_ws, size_t ws_size,
                              hipStream_t stream) {
  const float* in = (const float*)d_in[0];
  float* out = (float*)d_out;
  const int rows = out_size / 1024;          // 16384
  const int blocks = (rows + 7) / 8;         // 8 rows (waves) per 256-thread block
  nbdt_rules_kernel<<<dim3(blocks), dim3(256), 0, stream>>>(in, out, rows);
}